// EntropyFINQ_78091095375951
// MI455X (gfx1250) — compile-verified
//
#include <hip/hip_runtime.h>

#define ROWS   8192
#define COLS   16384
#define TPB    256
#define NBINS  11
#define FEPS   1e-8f

#define CHUNK_FLOATS 4096
#define NCHUNK       (COLS / CHUNK_FLOATS)          /* 4 */
#define F4_PER_CHUNK (CHUNK_FLOATS / 4 / TPB)       /* 4 per thread */

#if defined(__HIP_DEVICE_COMPILE__) && defined(__gfx1250__) && \
    __has_builtin(__builtin_amdgcn_global_load_async_to_lds_b128)
#define USE_ASYNC 1
#else
#define USE_ASYNC 0
#endif

#if USE_ASYNC
typedef int v4i_t __attribute__((ext_vector_type(4)));
typedef __attribute__((address_space(1))) v4i_t* gas_v4i_ptr;  // global int4*
typedef __attribute__((address_space(3))) v4i_t* las_v4i_ptr;  // LDS int4*
#endif

static_assert(F4_PER_CHUNK == 4, "manual unroll below assumes 4 issues/chunk");

// Monotonic float <-> uint key mapping: order-preserving, so min/max over keys
// equals min/max over floats; allows deterministic u32 atomics.
__device__ __forceinline__ unsigned fkey(float f) {
    unsigned u = __float_as_uint(f);
    return (u & 0x80000000u) ? ~u : (u | 0x80000000u);
}
__device__ __forceinline__ float keyf(unsigned k) {
    return (k & 0x80000000u) ? __uint_as_float(k & 0x7FFFFFFFu)
                             : __uint_as_float(~k);
}

// Stage one 16 KiB chunk of the row into LDS. ISA: INST_OFFSET is added to
// both the LDS and the global address, and our intra-chunk offsets match on
// both sides, so one base address + literal immediate offsets covers all 4
// issues (offset must be an integer constant expression at the call site).
__device__ __forceinline__ void issue_chunk(const float* __restrict__ g,
                                            float* l, int tid) {
#if USE_ASYNC
    gas_v4i_ptr gp = (gas_v4i_ptr)(g + tid * 4);
    las_v4i_ptr lp = (las_v4i_ptr)(l + tid * 4);
    __builtin_amdgcn_global_load_async_to_lds_b128(gp, lp, 0,     0);
    __builtin_amdgcn_global_load_async_to_lds_b128(gp, lp, 4096,  0);
    __builtin_amdgcn_global_load_async_to_lds_b128(gp, lp, 8192,  0);
    __builtin_amdgcn_global_load_async_to_lds_b128(gp, lp, 12288, 0);
#else
    const float4* g4 = (const float4*)g;
    float4* l4 = (float4*)l;
#pragma unroll
    for (int c = 0; c < F4_PER_CHUNK; ++c) {
        const int f4 = c * TPB + tid;
        l4[f4] = g4[f4];
    }
#endif
}

__device__ __forceinline__ void wait_async0() {
#if USE_ASYNC
#if __has_builtin(__builtin_amdgcn_s_wait_asynccnt)
    __builtin_amdgcn_s_wait_asynccnt(0);
#else
    asm volatile("s_wait_asynccnt 0" ::: "memory");
#endif
#endif
}

// Defined FIRST so the disasm snippet shows the hot loop.
__global__ void __launch_bounds__(TPB)
entropy_kernel(const float* __restrict__ x, const float* __restrict__ qp,
               const unsigned* __restrict__ ws, float* __restrict__ out) {
    __shared__ float sbuf[2][CHUNK_FLOATS];  // 2 x 16 KiB double buffer
    __shared__ unsigned hist[16];
    const int tid = threadIdx.x;
    if (tid < NBINS) hist[tid] = 0u;

    const float mn    = keyf(ws[0]);
    const float mx    = keyf(ws[1]);
    const float scale = 10.0f / (mx - mn + FEPS);
    const float bias  = -mn * scale;

    const float* rowp = x + (size_t)blockIdx.x * COLS;

    issue_chunk(rowp, sbuf[0], tid);

    // Packed per-thread counters: bin b -> byte (b&3) of reg (b>>2).
    // Max 64 elements/thread so 8-bit fields cannot overflow.
    unsigned c0 = 0, c1 = 0, c2 = 0;
    for (int k = 0; k < NCHUNK; ++k) {
        wait_async0();
        __syncthreads();  // chunk k fully in LDS; prev buffer reads done
        if (k + 1 < NCHUNK)
            issue_chunk(rowp + (k + 1) * CHUNK_FLOATS, sbuf[(k + 1) & 1], tid);
        const float4* s4 = (const float4*)sbuf[k & 1];
#pragma unroll
        for (int c = 0; c < F4_PER_CHUNK; ++c) {
            float4 v = s4[c * TPB + tid];  // consecutive lanes -> no bank conflicts
            float e[4] = {v.x, v.y, v.z, v.w};
#pragma unroll
            for (int j = 0; j < 4; ++j) {
                int b = (int)rintf(fmaf(e[j], scale, bias));  // v_rndne = jnp.round
                b = (b < 0) ? 0 : ((b > NBINS - 1) ? NBINS - 1 : b);
                const unsigned inc = 1u << ((b & 3) << 3);
                const unsigned hi  = (unsigned)b >> 2;
                c0 += (hi == 0u) ? inc : 0u;
                c1 += (hi == 1u) ? inc : 0u;
                c2 += (hi == 2u) ? inc : 0u;
            }
        }
    }

    // Merge packed counters into the block histogram (integer adds: deterministic).
#pragma unroll
    for (int b = 0; b < NBINS; ++b) {
        const unsigned reg = (b < 4) ? c0 : ((b < 8) ? c1 : c2);
        const unsigned cnt = (reg >> ((b & 3) * 8)) & 0xFFu;
        if (cnt) atomicAdd(&hist[b], cnt);
    }
    __syncthreads();

    if (tid == 0) {
        const float q = qp[0];
        float kpresent = 0.0f;
#pragma unroll
        for (int b = 0; b < NBINS; ++b) kpresent += (hist[b] > 0u) ? 1.0f : 0.0f;
        const float denom = (float)COLS + kpresent * FEPS;
        float ts = 0.0f;
#pragma unroll
        for (int b = 0; b < NBINS; ++b) {
            if (hist[b] > 0u) {
                const float p = ((float)hist[b] + FEPS) / denom;
                ts += powf(p, q);
            }
        }
        out[blockIdx.x] = (1.0f - ts) / (q - 1.0f + FEPS);
    }
}

__global__ void __launch_bounds__(TPB)
minmax_kernel(const float4* __restrict__ x, unsigned* __restrict__ ws) {
    __shared__ unsigned smin[TPB];
    __shared__ unsigned smax[TPB];
    const int tid = threadIdx.x;
    const size_t base   = (size_t)blockIdx.x * TPB + tid;
    const size_t stride = (size_t)gridDim.x * TPB;
    float mn = __builtin_inff();
    float mx = -__builtin_inff();
#pragma unroll 4
    for (int c = 0; c < 32; ++c) {
        float4 v = x[base + (size_t)c * stride];
        mn = fminf(mn, fminf(fminf(v.x, v.y), fminf(v.z, v.w)));
        mx = fmaxf(mx, fmaxf(fmaxf(v.x, v.y), fmaxf(v.z, v.w)));
    }
    smin[tid] = fkey(mn);
    smax[tid] = fkey(mx);
    __syncthreads();
    for (int s = TPB / 2; s > 0; s >>= 1) {
        if (tid < s) {
            unsigned a = smin[tid], b = smin[tid + s];
            smin[tid] = (a < b) ? a : b;
            unsigned c = smax[tid], d = smax[tid + s];
            smax[tid] = (c > d) ? c : d;
        }
        __syncthreads();
    }
    if (tid == 0) {
        atomicMin(&ws[0], smin[0]);
        atomicMax(&ws[1], smax[0]);
    }
}

__global__ void init_ws_kernel(unsigned* ws) {
    ws[0] = 0xFFFFFFFFu;  // running min key
    ws[1] = 0u;           // running max key
}

extern "C" void kernel_launch(void* const* d_in, const int* in_sizes, int n_in,
                              void* d_out, int out_size, void* d_ws, size_t ws_size,
                              hipStream_t stream) {
    (void)in_sizes; (void)n_in; (void)out_size; (void)ws_size;
    const float* x = (const float*)d_in[0];
    const float* q = (const float*)d_in[1];
    // d_in[2] = kappa, unused by the forward pass.
    float* out   = (float*)d_out;
    unsigned* ws = (unsigned*)d_ws;

    init_ws_kernel<<<dim3(1), dim3(1), 0, stream>>>(ws);
    minmax_kernel<<<dim3(4096), dim3(TPB), 0, stream>>>((const float4*)x, ws);
    entropy_kernel<<<dim3(ROWS), dim3(TPB), 0, stream>>>(x, q, ws, out);
}